// GCNModel_64527588655567
// MI455X (gfx1250) — compile-verified
//
#include <hip/hip_runtime.h>

// Problem geometry (fixed by the reference)
#define J_COLS 120000   // (N*M=16) * (T=300) * (V=25) columns of every activation matrix
#define NT_TOT 4800     // 16 * 300 spatial sites
#define VJ 25

typedef __attribute__((ext_vector_type(2))) float v2f;
typedef __attribute__((ext_vector_type(8))) float v8f;

// ---------------------------------------------------------------------------
// Pack weights (cout*3, cin) -> (cout*3, cinP) zero-padded so K-dim % 4 == 0
// ---------------------------------------------------------------------------
__global__ __launch_bounds__(256) void pack_w_kernel(const float* __restrict__ src,
                                                     float* __restrict__ dst,
                                                     int cin, int cinP, int n) {
  int idx = blockIdx.x * 256 + threadIdx.x;
  if (idx >= n) return;
  int r = idx / cinP;
  int c = idx - r * cinP;
  dst[idx] = (c < cin) ? src[r * cin + c] : 0.0f;
}

// ---------------------------------------------------------------------------
// Training-mode BatchNorm1d statistics: one block per channel (v*3+c), 75 ch,
// mean/var over 16 * 300 = 4800 samples.
// ---------------------------------------------------------------------------
__global__ __launch_bounds__(256) void bn_stats_kernel(const float* __restrict__ x,
                                                       float* __restrict__ stats) {
  const int ch = blockIdx.x;          // 0..74  == v*3 + c
  const int v = ch / 3, c = ch - 3 * v;
  const int tid = threadIdx.x;
  __shared__ float s1[256], s2[256];
  float sum = 0.0f, sq = 0.0f;
  for (int i = tid; i < 4800; i += 256) {
    int nm = i / 300, t = i - nm * 300;
    int n = nm >> 1, m = nm & 1;
    float val = x[(((n * 3 + c) * 300 + t) * 25 + v) * 2 + m];
    sum += val; sq += val * val;
  }
  s1[tid] = sum; s2[tid] = sq;
  __syncthreads();
  for (int off = 128; off > 0; off >>= 1) {
    if (tid < off) { s1[tid] += s1[tid + off]; s2[tid] += s2[tid + off]; }
    __syncthreads();
  }
  if (tid == 0) {
    float mean = s1[0] * (1.0f / 4800.0f);
    float var  = s2[0] * (1.0f / 4800.0f) - mean * mean;
    stats[2 * ch]     = mean;
    stats[2 * ch + 1] = rsqrtf(var + 1e-5f);
  }
}

// ---------------------------------------------------------------------------
// Apply BN and lay out H0 as [cinP=4][J_COLS]; row 3 is the zero K-pad row.
// ---------------------------------------------------------------------------
__global__ __launch_bounds__(256) void bn_apply_kernel(const float* __restrict__ x,
                                                       const float* __restrict__ gamma,
                                                       const float* __restrict__ beta,
                                                       const float* __restrict__ stats,
                                                       float* __restrict__ h0) {
  int idx = blockIdx.x * 256 + threadIdx.x;
  if (idx >= 4 * J_COLS) return;
  int c = idx / J_COLS;
  int j = idx - c * J_COLS;
  if (c == 3) { h0[idx] = 0.0f; return; }
  int v = j % 25;
  int rem = j / 25;
  int t = rem % 300;
  int nm = rem / 300;
  int n = nm >> 1, m = nm & 1;
  int ch = v * 3 + c;
  float val = x[(((n * 3 + c) * 300 + t) * 25 + v) * 2 + m];
  h0[idx] = (val - stats[2 * ch]) * stats[2 * ch + 1] * gamma[ch] + beta[ch];
}

// ---------------------------------------------------------------------------
// Fused s_gcn layer: per (nm,t) site, GEMM (W[3*cout,cin] x H[cin,25]) on the
// fp32 WMMA pipe, Y kept in LDS, then graph aggregation over (k,v) + ReLU.
//
// Activation window LDS layout (paired-K): element (c,v) lives at dword
//   (c>>1)*96 + 2*v + (c&1)
// so a B-fragment {K=ck, K=ck+1} at column v is ONE aligned ds_load_b64 into
// an even VGPR pair (WMMA source requirement), and the 96-dword (==32 mod 64)
// pair-stride puts the two lane-halves of a wave on disjoint bank sets.
// The window is filled with GLOBAL_LOAD_ASYNC_TO_LDS_B32 (ASYNCcnt path).
// ---------------------------------------------------------------------------
__global__ __launch_bounds__(256) void sgcn_layer_kernel(
    const float* __restrict__ Hin, float* __restrict__ Hout,
    const float* __restrict__ Wp, const float* __restrict__ bias,
    const float* __restrict__ Ag,
    int cinP, int cout, int Mrows, int finalFlag) {
  extern __shared__ float smem[];
  float* sA = smem;                    // 1880 floats (3*25*25 = 1875 used)
  float* sH = smem + 1880;             // cinP * 48  (paired-K layout)
  float* sY = sH + cinP * 48;          // Mrows * 26

  const int nt   = blockIdx.x;         // nm*300 + t
  const int tid  = threadIdx.x;
  const int wave = tid >> 5;
  const int lane = tid & 31;
  const int l16  = lane & 15;
  const int kh   = lane >> 4;          // which K-half of the wave this lane is in

  // Stage adjacency
  for (int i = tid; i < 1875; i += 256) sA[i] = Ag[i];

  // Stage the 25-column activation window via the async global->LDS pipe;
  // zero-fill pad columns v=25..31 with plain DS stores. Dynamic LDS starts
  // at offset 0 in this kernel (no static __shared__), so byte offsets are
  // computed directly.
  const unsigned sHbyteBase = 1880u * 4u;
  for (int i = tid; i < cinP * 32; i += 256) {
    int c = i >> 5, v = i & 31;
    int dstDword = (c >> 1) * 96 + 2 * v + (c & 1);
    if (v < VJ) {
      unsigned ldsOff = sHbyteBase + (unsigned)dstDword * 4u;
      const float* gp = Hin + (size_t)c * J_COLS + (size_t)nt * VJ + v;
      asm volatile("global_load_async_to_lds_b32 %0, %1, off"
                   :: "v"(ldsOff), "v"(gp)
                   : "memory");
    } else {
      sH[dstDword] = 0.0f;
    }
  }
  asm volatile("s_wait_asynccnt 0x0" ::: "memory");
  __syncthreads();

  // -------- Phase 1: WMMA GEMM, each wave owns 16(M) x 32(N) strips --------
  const v2f* sH2 = (const v2f*)sH;
  const int numMt = Mrows >> 4;
  for (int mt = wave; mt < numMt; mt += 8) {
    v8f acc0 = {};   // columns 0..15
    v8f acc1 = {};   // columns 16..31 (only 16..24 survive)
    const float* wrow = Wp + (mt * 16 + l16) * cinP + 2 * kh;
    for (int kk = 0; kk < cinP; kk += 4) {
      v2f a;                             // A 16x4: lanes<16 K=kk..kk+1, lanes>=16 K=kk+2..kk+3
      a[0] = wrow[kk];
      a[1] = wrow[kk + 1];
      const v2f* hb = sH2 + ((kk >> 1) + kh) * 48 + l16;
      v2f b0 = hb[0];                    // {H[ck][l16],    H[ck+1][l16]}
      v2f b1 = hb[16];                   // {H[ck][16+l16], H[ck+1][16+l16]}
      acc0 = __builtin_amdgcn_wmma_f32_16x16x4_f32(false, a, false, b0,
                                                   (short)0, acc0, false, false);
      acc1 = __builtin_amdgcn_wmma_f32_16x16x4_f32(false, a, false, b1,
                                                   (short)0, acc1, false, false);
    }
    // D layout: VGPR r -> (M=r, N=l16) for lanes<16, (M=r+8, N=l16) for lanes>=16
#pragma unroll
    for (int r = 0; r < 8; ++r) {
      int row = mt * 16 + r + 8 * kh;
      float bv = bias[row];
      sY[row * 26 + l16] = acc0[r] + bv;
      if (16 + l16 < VJ) sY[row * 26 + 16 + l16] = acc1[r] + bv;
    }
  }
  __syncthreads();

  // -------- Phase 2: graph aggregation h'[c,w] = relu(sum_{k,v} Y A) -------
  const int total = cout * VJ;
  for (int idx = tid; idx < total; idx += 256) {
    int c = idx / VJ;
    int w = idx - c * VJ;
    float acc = 0.0f;
#pragma unroll
    for (int k = 0; k < 3; ++k) {
      const float* yrow = sY + (k * cout + c) * 26;
      const float* arow = sA + k * 625 + w;
#pragma unroll
      for (int v = 0; v < VJ; ++v) acc = fmaf(yrow[v], arow[v * 25], acc);
    }
    acc = fmaxf(acc, 0.0f);
    if (finalFlag) Hout[(nt * cout + c) * VJ + w] = acc;   // final flat layout
    else           Hout[c * J_COLS + nt * VJ + w] = acc;   // next layer's input
  }
}

// ---------------------------------------------------------------------------
// Host driver
// ---------------------------------------------------------------------------
extern "C" void kernel_launch(void* const* d_in, const int* in_sizes, int n_in,
                              void* d_out, int out_size, void* d_ws, size_t ws_size,
                              hipStream_t stream) {
  (void)in_sizes; (void)n_in; (void)out_size; (void)ws_size;
  const float* x     = (const float*)d_in[0];
  const float* A     = (const float*)d_in[1];
  const float* gamma = (const float*)d_in[2];
  const float* beta  = (const float*)d_in[3];

  static const int CIN[10]  = {3, 64, 64, 64, 64, 128, 128, 128, 256, 256};
  static const int COUT[10] = {64, 64, 64, 64, 128, 128, 128, 256, 256, 256};

  float* wsf = (float*)d_ws;
  // ws layout (floats): [0,150) BN stats | packed weights | two ping-pong
  // activation buffers of 256 x 120000 each (~123 MB apiece).
  size_t off = 160;
  size_t offW[10];
  for (int l = 0; l < 10; ++l) {
    int cinP = (CIN[l] + 3) & ~3;
    offW[l] = off;
    off += (size_t)(COUT[l] * 3) * (size_t)cinP;
  }
  off = (off + 15) & ~(size_t)15;
  float* buf0 = wsf + off;
  float* buf1 = buf0 + (size_t)256 * J_COLS;

  for (int l = 0; l < 10; ++l) {
    int cinP = (CIN[l] + 3) & ~3;
    int n = COUT[l] * 3 * cinP;
    pack_w_kernel<<<(n + 255) / 256, 256, 0, stream>>>(
        (const float*)d_in[4 + l], wsf + offW[l], CIN[l], cinP, n);
  }

  bn_stats_kernel<<<75, 256, 0, stream>>>(x, wsf);
  bn_apply_kernel<<<(4 * J_COLS + 255) / 256, 256, 0, stream>>>(x, gamma, beta, wsf, buf0);

  for (int l = 0; l < 10; ++l) {
    int cinP  = (CIN[l] + 3) & ~3;
    int Mrows = COUT[l] * 3;
    const float* src = (l & 1) ? buf1 : buf0;
    float* dst = (l == 9) ? (float*)d_out : ((l & 1) ? buf0 : buf1);
    size_t ldsBytes = (size_t)(1880 + cinP * 48 + Mrows * 26) * sizeof(float);
    sgcn_layer_kernel<<<NT_TOT, 256, ldsBytes, stream>>>(
        src, dst, wsf + offW[l], (const float*)d_in[14 + l], A,
        cinP, COUT[l], Mrows, (l == 9) ? 1 : 0);
  }
}